// AVWDCRNN2_48644799595031
// MI455X (gfx1250) — compile-verified
//
#include <hip/hip_runtime.h>
#include <math.h>

// Problem constants (from the reference)
#define NN    307   // nodes
#define NP    320   // nodes padded to multiple of 32
#define TT    12    // timesteps
#define BB    128   // batch
#define DOUTC 64
#define DEC   10

typedef __attribute__((ext_vector_type(2))) float v2f;
typedef __attribute__((ext_vector_type(8))) float v8f;

__device__ __forceinline__ v8f wmma4_f32(v2f a, v2f b, v8f c) {
  // D = A(16x4) * B(4x16) + C(16x16), fp32 — CDNA5 V_WMMA_F32_16X16X4_F32
  return __builtin_amdgcn_wmma_f32_16x16x4_f32(
      /*neg_a=*/false, a, /*neg_b=*/false, b,
      /*c_mod=*/(short)0, c, /*reuse_a=*/false, /*reuse_b=*/false);
}

// ---------------------------------------------------------------------------
// Precompute: per-node weights  Wn[n][k][ip][o] = sum_d E[n,d]*pool[d,k,i,o]
// (ip >= Ci positions are zero padding so K-tiles of 4 are always full)
// ---------------------------------------------------------------------------
__global__ void avw_make_W(const float* __restrict__ E, const float* __restrict__ pool,
                           float* __restrict__ Wn, int Ci, int CiP, int Co, int total) {
  int idx = blockIdx.x * blockDim.x + threadIdx.x;
  if (idx >= total) return;
  int o  = idx % Co;
  int ip = (idx / Co) % CiP;
  int k  = (idx / (Co * CiP)) % 2;
  int n  = idx / (Co * CiP * 2);
  float v = 0.f;
  if (ip < Ci) {
    const float* e = E + n * DEC;
#pragma unroll
    for (int d = 0; d < DEC; ++d)
      v += e[d] * pool[(((size_t)d * 2 + k) * Ci + ip) * Co + o];
  }
  Wn[idx] = v;
}

__global__ void avw_make_b(const float* __restrict__ E, const float* __restrict__ pool,
                           float* __restrict__ bn, int Co, int total) {
  int idx = blockIdx.x * blockDim.x + threadIdx.x;
  if (idx >= total) return;
  int o = idx % Co;
  int n = idx / Co;
  float v = 0.f;
#pragma unroll
  for (int d = 0; d < DEC; ++d) v += E[n * DEC + d] * pool[d * Co + o];
  bn[idx] = v;
}

// Transposed zero-padded adjacency: AT[m][n] = adj[n][m]  (coalesced A-frag loads)
__global__ void avw_make_apadT(const float* __restrict__ adj, float* __restrict__ AT) {
  int idx = blockIdx.x * blockDim.x + threadIdx.x;
  if (idx >= NP * NP) return;
  int n = idx % NP, m = idx / NP;            // AT[m*NP + n]
  AT[idx] = (n < NN && m < NN) ? adj[(size_t)n * NN + m] : 0.f;
}

// h[l] layout [n][b][o]  <-  init_state [L,B,N,DOUT]
__global__ void avw_init_h(const float* __restrict__ s, float* __restrict__ h, int l) {
  int idx = blockIdx.x * blockDim.x + threadIdx.x;
  if (idx >= NN * BB * DOUTC) return;
  int o = idx % DOUTC;
  int b = (idx / DOUTC) % BB;
  int n = idx / (DOUTC * BB);
  h[idx] = s[(((size_t)l * BB + b) * NN + n) * DOUTC + o];
}

// ---------------------------------------------------------------------------
// cat builders: cat layout [m (padded NP)][b][c (padded CiP)]
// ---------------------------------------------------------------------------
__global__ void avw_cat_l0(const float* __restrict__ x, const float* __restrict__ h0,
                           const float* __restrict__ zr, float* __restrict__ cat,
                           int t, int use_z) {
  const int CiP = 68;
  int idx = blockIdx.x * blockDim.x + threadIdx.x;
  if (idx >= NP * BB * CiP) return;
  int c = idx % CiP;
  int b = (idx / CiP) % BB;
  int m = idx / (CiP * BB);
  float v = 0.f;
  if (m < NN) {
    if (c < 2) {
      v = x[(((size_t)b * TT + t) * NN + m) * 2 + c];            // x[B,T,N,DIN]
    } else if (c < 66) {
      float hv = h0[((size_t)m * BB + b) * DOUTC + (c - 2)];
      if (use_z) hv *= zr[((size_t)m * BB + b) * 128 + (c - 2)]; // z = zr[...,0:64]
      v = hv;
    }
  }
  cat[idx] = v;
}

__global__ void avw_cat_l1(const float* __restrict__ h0, const float* __restrict__ h1,
                           const float* __restrict__ zr, float* __restrict__ cat,
                           int use_z) {
  const int CiP = 128;
  int idx = blockIdx.x * blockDim.x + threadIdx.x;
  if (idx >= NP * BB * CiP) return;
  int c = idx % CiP;
  int b = (idx / CiP) % BB;
  int m = idx / (CiP * BB);
  float v = 0.f;
  if (m < NN) {
    if (c < 64) {
      v = h0[((size_t)m * BB + b) * DOUTC + c];                  // xt = layer-0 h(t)
    } else {
      float hv = h1[((size_t)m * BB + b) * DOUTC + (c - 64)];
      if (use_z) hv *= zr[((size_t)m * BB + b) * 128 + (c - 64)];
      v = hv;
    }
  }
  cat[idx] = v;
}

// ---------------------------------------------------------------------------
// Graph conv GEMM:  Y[NP, NC] = A[NP, NP] @ X[NP, NC],   NC = B * CiP
// A supplied transposed (AT[m][n]).  One 32x64 output tile per wave:
// 2 (M) x 4 (N) register tiles, 8 v8f accumulators, 8 WMMAs per K-step of 4.
// ---------------------------------------------------------------------------
__global__ void avw_graph_gemm(const float* __restrict__ AT, const float* __restrict__ X,
                               float* __restrict__ Y, int NC, int cstrips, int ntiles) {
  int wave = threadIdx.x >> 5;
  int lane = threadIdx.x & 31;
  int tile = blockIdx.x * (blockDim.x >> 5) + wave;
  if (tile >= ntiles) return;                 // whole-wave exit: EXEC stays full
  int rt = tile / cstrips, ct = tile % cstrips;
  int row0 = rt << 5, col0 = ct << 6;
  int lh = lane >> 4, l15 = lane & 15;

  v8f acc[2][4] = {};
  const float* ap = AT + row0 + l15;          // + ka*NP + 16*i
  const float* bp = X + col0 + l15;           // + ka*NC + 16*j
#pragma unroll 2
  for (int k0 = 0; k0 < NP; k0 += 4) {
    int ka = k0 + 2 * lh;                     // lanes16-31 take K=k0+2,k0+3
    v2f a[2], b[4];
#pragma unroll
    for (int i = 0; i < 2; ++i) {
      a[i].x = ap[(size_t)ka * NP + 16 * i];
      a[i].y = ap[(size_t)(ka + 1) * NP + 16 * i];
    }
#pragma unroll
    for (int j = 0; j < 4; ++j) {
      b[j].x = bp[(size_t)ka * NC + 16 * j];
      b[j].y = bp[(size_t)(ka + 1) * NC + 16 * j];
    }
#pragma unroll
    for (int i = 0; i < 2; ++i)
#pragma unroll
      for (int j = 0; j < 4; ++j)
        acc[i][j] = wmma4_f32(a[i], b[j], acc[i][j]);
  }
#pragma unroll
  for (int i = 0; i < 2; ++i)
#pragma unroll
    for (int j = 0; j < 4; ++j) {
      float* yp = Y + (size_t)(row0 + 16 * i + 8 * lh) * NC + col0 + 16 * j + l15;
#pragma unroll
      for (int r = 0; r < 8; ++r) yp[(size_t)r * NC] = acc[i][j][r];
    }
}

// ---------------------------------------------------------------------------
// Per-node GEMM + bias + activation:
//   out[n][b][o] = act( sum_i Xc[n,b,i]*W[n,0,i,o] + Xa[n,b,i]*W[n,1,i,o] + bn[n,o] )
// M = B = 128 (rows b), Ndim = Co, K = 2*CiP.  32x64 tile/wave.  act:0=sigmoid 1=tanh
// ---------------------------------------------------------------------------
__global__ void avw_node_gemm(const float* __restrict__ Xc, const float* __restrict__ Xa,
                              const float* __restrict__ Wn, const float* __restrict__ bn,
                              float* __restrict__ out, int CiP, int Co, int act, int ntiles) {
  int wave = threadIdx.x >> 5;
  int lane = threadIdx.x & 31;
  int tile = blockIdx.x * (blockDim.x >> 5) + wave;
  if (tile >= ntiles) return;
  int cstrips = Co >> 6;                      // 64-wide column strips (2 or 1)
  int per_n = (BB >> 5) * cstrips;
  int n = tile / per_n;
  int rem = tile - n * per_n;
  int rt = rem / cstrips, ct = rem - rt * cstrips;
  int row0 = rt << 5, col0 = ct << 6;
  int lh = lane >> 4, l15 = lane & 15;

  size_t xoff[2];
#pragma unroll
  for (int i = 0; i < 2; ++i)
    xoff[i] = ((size_t)n * BB + row0 + 16 * i + l15) * CiP;
  const float* w = Wn + (size_t)n * 2 * CiP * Co + col0 + l15;  // + ka*Co + 16*j

  v8f acc[2][4] = {};
  int Kt = 2 * CiP;                           // CiP multiple of 4 by construction
  for (int k0 = 0; k0 < Kt; k0 += 4) {
    const float* xb = (k0 < CiP) ? Xc : Xa;   // k=0 support (identity) vs k=1 (A@cat)
    int kl = ((k0 < CiP) ? k0 : k0 - CiP) + 2 * lh;
    int ka = k0 + 2 * lh;
    if ((k0 & 31) == 0 && k0 + 32 < Kt)       // wave-uniform: prefetch next W panel
      __builtin_prefetch(w + (size_t)(k0 + 32) * Co, 0, 1);
    v2f a[2], b[4];
#pragma unroll
    for (int i = 0; i < 2; ++i) {
      a[i].x = xb[xoff[i] + kl];
      a[i].y = xb[xoff[i] + kl + 1];
    }
#pragma unroll
    for (int j = 0; j < 4; ++j) {
      b[j].x = w[(size_t)ka * Co + 16 * j];
      b[j].y = w[(size_t)(ka + 1) * Co + 16 * j];
    }
#pragma unroll
    for (int i = 0; i < 2; ++i)
#pragma unroll
      for (int j = 0; j < 4; ++j)
        acc[i][j] = wmma4_f32(a[i], b[j], acc[i][j]);
  }
#pragma unroll
  for (int j = 0; j < 4; ++j) {
    float bias = bn[(size_t)n * Co + col0 + 16 * j + l15];
#pragma unroll
    for (int i = 0; i < 2; ++i) {
      float* op = out + ((size_t)n * BB + row0 + 16 * i + 8 * lh) * Co + col0 + 16 * j + l15;
#pragma unroll
      for (int r = 0; r < 8; ++r) {
        float v = acc[i][j][r] + bias;
        v = act ? tanhf(v) : (1.f / (1.f + __expf(-v)));
        op[(size_t)r * Co] = v;
      }
    }
  }
}

// h' = r*h + (1-r)*hc ; optionally scatter into d_out current[B,T,N,DOUT]
__global__ void avw_gru_update(const float* __restrict__ zr, const float* __restrict__ hc,
                               float* __restrict__ h, float* __restrict__ outc,
                               int t, int write_out) {
  int idx = blockIdx.x * blockDim.x + threadIdx.x;
  if (idx >= NN * BB * DOUTC) return;
  int o = idx % DOUTC;
  int b = (idx / DOUTC) % BB;
  int n = idx / (DOUTC * BB);
  float r  = zr[((size_t)n * BB + b) * 128 + 64 + o];   // r = zr[...,64:128]
  float hn = r * h[idx] + (1.f - r) * hc[idx];
  h[idx] = hn;
  if (write_out)
    outc[(((size_t)b * TT + t) * NN + n) * DOUTC + o] = hn;
}

// last_states[l] : dst[b][n][o] = h[n][b][o]
__global__ void avw_copy_last(const float* __restrict__ h, float* __restrict__ dst) {
  int idx = blockIdx.x * blockDim.x + threadIdx.x;
  if (idx >= NN * BB * DOUTC) return;
  int o = idx % DOUTC;
  int b = (idx / DOUTC) % BB;
  int n = idx / (DOUTC * BB);
  dst[((size_t)b * NN + n) * DOUTC + o] = h[idx];
}

// ---------------------------------------------------------------------------
extern "C" void kernel_launch(void* const* d_in, const int* in_sizes, int n_in,
                              void* d_out, int out_size, void* d_ws, size_t ws_size,
                              hipStream_t stream) {
  (void)in_sizes; (void)n_in; (void)out_size; (void)ws_size;
  const float* x    = (const float*)d_in[0];
  const float* st0  = (const float*)d_in[1];
  const float* E    = (const float*)d_in[2];
  const float* adj  = (const float*)d_in[3];
  const float* wg0  = (const float*)d_in[4];
  const float* bg0  = (const float*)d_in[5];
  const float* wu0  = (const float*)d_in[6];
  const float* bu0  = (const float*)d_in[7];
  const float* wg1  = (const float*)d_in[8];
  const float* bg1  = (const float*)d_in[9];
  const float* wu1  = (const float*)d_in[10];
  const float* bu1  = (const float*)d_in[11];
  float* out = (float*)d_out;

  // Workspace carve-up (floats)
  float* ws = (float*)d_ws;
  size_t off = 0;
  auto take = [&](size_t nflt) { float* p = ws + off; off += nflt; return p; };
  const int CiP0 = 68, CiP1 = 128;
  float* Wg0n = take((size_t)NN * 2 * CiP0 * 128);
  float* Wu0n = take((size_t)NN * 2 * CiP0 * 64);
  float* Wg1n = take((size_t)NN * 2 * CiP1 * 128);
  float* Wu1n = take((size_t)NN * 2 * CiP1 * 64);
  float* bg0n = take((size_t)NN * 128);
  float* bu0n = take((size_t)NN * 64);
  float* bg1n = take((size_t)NN * 128);
  float* bu1n = take((size_t)NN * 64);
  float* ApT  = take((size_t)NP * NP);
  float* catb = take((size_t)NP * BB * 128);
  float* aggb = take((size_t)NP * BB * 128);
  float* zrb  = take((size_t)NN * BB * 128);
  float* hcb  = take((size_t)NN * BB * 64);
  float* h0   = take((size_t)NN * BB * 64);
  float* h1   = take((size_t)NN * BB * 64);

  auto grid1 = [](int total) { return dim3((total + 255) / 256); };
  const dim3 thr(256);

  // ---- one-time (per launch) precompute ----
  {
    int tg0 = NN * 2 * CiP0 * 128, tu0 = NN * 2 * CiP0 * 64;
    int tg1 = NN * 2 * CiP1 * 128, tu1 = NN * 2 * CiP1 * 64;
    avw_make_W<<<grid1(tg0), thr, 0, stream>>>(E, wg0, Wg0n, 66, CiP0, 128, tg0);
    avw_make_W<<<grid1(tu0), thr, 0, stream>>>(E, wu0, Wu0n, 66, CiP0, 64,  tu0);
    avw_make_W<<<grid1(tg1), thr, 0, stream>>>(E, wg1, Wg1n, 128, CiP1, 128, tg1);
    avw_make_W<<<grid1(tu1), thr, 0, stream>>>(E, wu1, Wu1n, 128, CiP1, 64,  tu1);
    avw_make_b<<<grid1(NN * 128), thr, 0, stream>>>(E, bg0, bg0n, 128, NN * 128);
    avw_make_b<<<grid1(NN * 64),  thr, 0, stream>>>(E, bu0, bu0n, 64,  NN * 64);
    avw_make_b<<<grid1(NN * 128), thr, 0, stream>>>(E, bg1, bg1n, 128, NN * 128);
    avw_make_b<<<grid1(NN * 64),  thr, 0, stream>>>(E, bu1, bu1n, 64,  NN * 64);
    avw_make_apadT<<<grid1(NP * NP), thr, 0, stream>>>(adj, ApT);
    avw_init_h<<<grid1(NN * BB * 64), thr, 0, stream>>>(st0, h0, 0);
    avw_init_h<<<grid1(NN * BB * 64), thr, 0, stream>>>(st0, h1, 1);
  }

  // GEMM tile counts (per wave, 32x64 tiles; 4 waves / 128-thread block)
  const int NC0 = BB * CiP0, NC1 = BB * CiP1;           // 8704, 16384
  const int cs0 = NC0 / 64, cs1 = NC1 / 64;             // 136, 256
  const int gt0 = (NP / 32) * cs0;                      // 1360
  const int gt1 = (NP / 32) * cs1;                      // 2560
  const int ngate = NN * (BB / 32) * (128 / 64);        // 2456
  const int nupd  = NN * (BB / 32) * (64 / 64);         // 1228
  auto gridw = [](int tiles) { return dim3((tiles + 3) / 4); };
  const dim3 thrw(128);
  const int eN = NN * BB * 64;

  for (int t = 0; t < TT; ++t) {
    // ----- layer 0 cell -----
    avw_cat_l0<<<grid1(NP * BB * CiP0), thr, 0, stream>>>(x, h0, nullptr, catb, t, 0);
    avw_graph_gemm<<<gridw(gt0), thrw, 0, stream>>>(ApT, catb, aggb, NC0, cs0, gt0);
    avw_node_gemm<<<gridw(ngate), thrw, 0, stream>>>(catb, aggb, Wg0n, bg0n, zrb, CiP0, 128, 0, ngate);
    avw_cat_l0<<<grid1(NP * BB * CiP0), thr, 0, stream>>>(x, h0, zrb, catb, t, 1);
    avw_graph_gemm<<<gridw(gt0), thrw, 0, stream>>>(ApT, catb, aggb, NC0, cs0, gt0);
    avw_node_gemm<<<gridw(nupd), thrw, 0, stream>>>(catb, aggb, Wu0n, bu0n, hcb, CiP0, 64, 1, nupd);
    avw_gru_update<<<grid1(eN), thr, 0, stream>>>(zrb, hcb, h0, nullptr, t, 0);
    // ----- layer 1 cell (xt = h0 at time t) -----
    avw_cat_l1<<<grid1(NP * BB * CiP1), thr, 0, stream>>>(h0, h1, nullptr, catb, 0);
    avw_graph_gemm<<<gridw(gt1), thrw, 0, stream>>>(ApT, catb, aggb, NC1, cs1, gt1);
    avw_node_gemm<<<gridw(ngate), thrw, 0, stream>>>(catb, aggb, Wg1n, bg1n, zrb, CiP1, 128, 0, ngate);
    avw_cat_l1<<<grid1(NP * BB * CiP1), thr, 0, stream>>>(h0, h1, zrb, catb, 1);
    avw_graph_gemm<<<gridw(gt1), thrw, 0, stream>>>(ApT, catb, aggb, NC1, cs1, gt1);
    avw_node_gemm<<<gridw(nupd), thrw, 0, stream>>>(catb, aggb, Wu1n, bu1n, hcb, CiP1, 64, 1, nupd);
    avw_gru_update<<<grid1(eN), thr, 0, stream>>>(zrb, hcb, h1, out, t, 1);
  }

  // last_states [L,B,N,DOUT] appended after current [B,T,N,DOUT]
  float* last = out + (size_t)BB * TT * NN * DOUTC;
  avw_copy_last<<<grid1(eN), thr, 0, stream>>>(h0, last);
  avw_copy_last<<<grid1(eN), thr, 0, stream>>>(h1, last + (size_t)BB * NN * DOUTC);
}